// GraphSAGELinkPredictor_36464272343628
// MI455X (gfx1250) — compile-verified
//
#include <hip/hip_runtime.h>
#include <hip/hip_bf16.h>

// GraphSAGE 2-layer (mean aggregation) for MI455X / gfx1250.
// Pipeline: degree count -> edge scatter-add -> fused WMMA fp32 GEMM
//   out = relu_opt( (sum/deg) @ W_l + b + X @ W_r )
// using V_WMMA_F32_16X16X4_F32 with LDS-staged A tiles (conflict-free padding).

typedef __attribute__((ext_vector_type(2))) float v2f;
typedef __attribute__((ext_vector_type(8))) float v8f;

#define IN_CH 128
#define LDS_PITCH (IN_CH + 4)   // 132 floats = 528B row stride:
                                // 16B-aligned rows, lane m -> bank 4m (no conflicts)

// ---------------------------------------------------------------- zero fill
__global__ void sage_zero_kernel(float* __restrict__ p, size_t n) {
    size_t i = (size_t)blockIdx.x * blockDim.x + threadIdx.x;
    size_t stride = (size_t)gridDim.x * blockDim.x;
    for (; i < n; i += stride) p[i] = 0.0f;
}

// ---------------------------------------------------------------- degree
__global__ void sage_degree_kernel(const int* __restrict__ dst,
                                   float* __restrict__ cnt, int E) {
    int e = blockIdx.x * blockDim.x + threadIdx.x;
    if (e < E) atomicAdd(&cnt[dst[e]], 1.0f);
}

// ---------------------------------------------------------------- scatter
// 32 lanes per edge; each lane gathers a float4 (coalesced 512B per edge)
// and does 4 fp32 atomic adds into the destination row (L2-resident).
__global__ void sage_scatter_kernel(const int* __restrict__ src,
                                    const int* __restrict__ dst,
                                    const float* __restrict__ feat,
                                    float* __restrict__ sum, int E) {
    int t = blockIdx.x * blockDim.x + threadIdx.x;
    int e = t >> 5;
    int lane = t & 31;
    if (e >= E) return;
    int s = src[e];
    int d = dst[e];
    const float4* sp = (const float4*)(feat + (size_t)s * IN_CH) + lane;
    float4 v = *sp;
    float* dp = sum + (size_t)d * IN_CH + lane * 4;
    atomicAdd(dp + 0, v.x);
    atomicAdd(dp + 1, v.y);
    atomicAdd(dp + 2, v.z);
    atomicAdd(dp + 3, v.w);
}

// ---------------------------------------------------------------- fused GEMM
// out[r, :] = relu_opt( (sum[r,:] / max(cnt[r],1)) @ Wl + bias + X[r,:] @ Wr )
// Block = 256 threads = 8 waves; one 16-row M-tile per block, one 16-col
// N-tile per wave. A tiles (normalized aggr + self) staged once in LDS and
// shared by all 8 waves; B fragments loaded from global (per-wave-unique,
// L2 resident). grid.x = n_rows / 16 (100000 = 6250 * 16, exact).
//
// fp32 WMMA fragment layouts (ISA 7.12.2):
//   A 16x4 : lanes 0-15 -> M=lane, VGPR{0,1}=K{0,1}; lanes 16-31 -> K{2,3}
//   B 4x16 : lanes 0-15 -> N=lane, VGPR{0,1}=K{0,1}; lanes 16-31 -> K{2,3}
//   C 16x16: VGPR r, lanes 0-15 -> (M=r, N=lane); lanes 16-31 -> (M=r+8)
__global__ void sage_gemm_kernel(const float* __restrict__ sum,
                                 const float* __restrict__ cnt,
                                 const float* __restrict__ X,
                                 const float* __restrict__ Wl,
                                 const float* __restrict__ Wr,
                                 const float* __restrict__ bias,
                                 float* __restrict__ out,
                                 int do_relu) {
    __shared__ float ldsA[16 * LDS_PITCH];   // normalized aggregate rows
    __shared__ float ldsX[16 * LDS_PITCH];   // self-feature rows

    const int tid  = threadIdx.x;
    const int row0 = blockIdx.x * 16;

    // ---- stage A tiles: 16 rows x 128 cols as float4 (512 vec4 / 256 thr) ----
    for (int i = tid; i < 16 * (IN_CH / 4); i += 256) {
        const int r  = i >> 5;           // / (IN_CH/4)
        const int c4 = i & 31;           // % (IN_CH/4)
        const float invc = 1.0f / fmaxf(cnt[row0 + r], 1.0f);
        const float4 s  = ((const float4*)(sum + (size_t)(row0 + r) * IN_CH))[c4];
        const float4 xv = ((const float4*)(X   + (size_t)(row0 + r) * IN_CH))[c4];
        float4 a;
        a.x = s.x * invc; a.y = s.y * invc; a.z = s.z * invc; a.w = s.w * invc;
        ((float4*)(ldsA + r * LDS_PITCH))[c4] = a;
        ((float4*)(ldsX + r * LDS_PITCH))[c4] = xv;
    }
    __syncthreads();

    const int wave = tid >> 5;                  // 0..7 -> column tile
    const int lane = tid & 31;
    const int col0 = wave * 16;
    const int mn   = lane & 15;                 // row (A) / col (B,C)
    const int koff = (lane >> 4) << 1;          // 0 or 2

    // accumulator initialized with bias (bias depends only on N = col)
    const float bv = bias[col0 + mn];
    v8f acc;
#pragma unroll
    for (int r = 0; r < 8; ++r) acc[r] = bv;

    const float* aRow = ldsA + mn * LDS_PITCH;
    const float* xRow = ldsX + mn * LDS_PITCH;

    for (int k = 0; k < IN_CH; k += 4) {
        // A fragments from LDS (ds_load_b64, conflict-free via pitch padding)
        v2f a_aggr = *(const v2f*)(aRow + k + koff);
        v2f a_self = *(const v2f*)(xRow + k + koff);

        // B fragments from weight matrices (row-major [128 x 128], L2 hot)
        v2f bl, br;
        bl[0] = Wl[(size_t)(k + koff + 0) * IN_CH + col0 + mn];
        bl[1] = Wl[(size_t)(k + koff + 1) * IN_CH + col0 + mn];
        br[0] = Wr[(size_t)(k + koff + 0) * IN_CH + col0 + mn];
        br[1] = Wr[(size_t)(k + koff + 1) * IN_CH + col0 + mn];

        acc = __builtin_amdgcn_wmma_f32_16x16x4_f32(
            false, a_aggr, false, bl, (short)0, acc, false, false);
        acc = __builtin_amdgcn_wmma_f32_16x16x4_f32(
            false, a_self, false, br, (short)0, acc, false, false);
    }

    const int rowBase = row0 + ((lane >> 4) << 3);   // +8 for lanes 16-31
#pragma unroll
    for (int r = 0; r < 8; ++r) {
        float v = acc[r];
        if (do_relu) v = fmaxf(v, 0.0f);
        out[(size_t)(rowBase + r) * IN_CH + col0 + mn] = v;
    }
}

// ---------------------------------------------------------------- launch
extern "C" void kernel_launch(void* const* d_in, const int* in_sizes, int n_in,
                              void* d_out, int out_size, void* d_ws, size_t ws_size,
                              hipStream_t stream) {
    const float* x   = (const float*)d_in[0];
    const int*  edge = (const int*)d_in[1];
    const float* Wl1 = (const float*)d_in[2];
    const float* b1  = (const float*)d_in[3];
    const float* Wr1 = (const float*)d_in[4];
    const float* Wl2 = (const float*)d_in[5];
    const float* b2  = (const float*)d_in[6];
    const float* Wr2 = (const float*)d_in[7];
    float* out = (float*)d_out;

    const int E = in_sizes[1] / 2;
    const int n = in_sizes[0] / IN_CH;          // 100000 (multiple of 16)
    const int* src = edge;
    const int* dst = edge + E;

    // Workspace layout (contiguous so the zeroed region is one span):
    //   cnt  [n] | sum1 [n*128] | sum2 [n*128] | h [n*128]
    float* ws   = (float*)d_ws;
    float* cnt  = ws;
    float* sum1 = cnt + n;
    float* sum2 = sum1 + (size_t)n * IN_CH;
    float* h    = sum2 + (size_t)n * IN_CH;

    const size_t zero_elems = (size_t)n + 2 * (size_t)n * IN_CH;
    sage_zero_kernel<<<2048, 256, 0, stream>>>(ws, zero_elems);

    sage_degree_kernel<<<(E + 255) / 256, 256, 0, stream>>>(dst, cnt, E);

    // ---- layer 1 ----
    {
        long long threads = (long long)E * 32;
        int blocks = (int)((threads + 255) / 256);
        sage_scatter_kernel<<<blocks, 256, 0, stream>>>(src, dst, x, sum1, E);
    }
    sage_gemm_kernel<<<n / 16, 256, 0, stream>>>(sum1, cnt, x, Wl1, Wr1, b1, h,
                                                 /*do_relu=*/1);

    // ---- layer 2 ----
    {
        long long threads = (long long)E * 32;
        int blocks = (int)((threads + 255) / 256);
        sage_scatter_kernel<<<blocks, 256, 0, stream>>>(src, dst, h, sum2, E);
    }
    sage_gemm_kernel<<<n / 16, 256, 0, stream>>>(sum2, cnt, h, Wl2, Wr2, b2, out,
                                                 /*do_relu=*/0);
}